// EEBasis_68728066670779
// MI455X (gfx1250) — compile-verified
//
#include <hip/hip_runtime.h>

typedef float v4f __attribute__((ext_vector_type(4)));

#define LOG2E 1.44269504088896340736f
#define ROWS_PER_TILE 16   // rows of N handled per block iteration
#define TPB 256            // 8 wave32 waves per block

// Use CDNA5 async global->LDS DMA for the stage phase if the toolchain
// exposes the builtin; otherwise fall back to NT load + ds_store.
#if defined(__HIP_DEVICE_COMPILE__) && __has_builtin(__builtin_amdgcn_global_load_async_to_lds_b32)
#define USE_ASYNC_LDS 1
#else
#define USE_ASYNC_LDS 0
#endif

#if USE_ASYNC_LDS
typedef __attribute__((address_space(1))) int* gptr_i32;
typedef __attribute__((address_space(3))) int* lptr_i32;
#endif

// out[n, s] = exp(-|zeta_s| * sqrt(diffs[n, cidx[s], 3])), out is [N, 256] fp32.
// Stage: copy the .w component for 16 rows x 64 centers into LDS
// (async DMA on CDNA5, ASYNCcnt-tracked). Compute: per thread, 4 shells ->
// 4x v_sqrt_f32 + 4x v_exp_f32 + one coalesced b128 NT store.
__global__ __launch_bounds__(TPB) void eebasis_kernel(
    const float* __restrict__ diffs,   // [N, 64, 4] fp32
    const float* __restrict__ zetas,   // [256] fp32
    const int*   __restrict__ cidx,    // [256] int32
    float* __restrict__ out,           // [N, 256] fp32
    int nrows)                          // N
{
    __shared__ __align__(16) float s_zl[256];          // |zeta| * log2(e)
    __shared__ int   s_ci[256];                        // shell -> center
    __shared__ float s_w[ROWS_PER_TILE * 64];          // raw d2 per (row, center)

    const int t = threadIdx.x;

    // one-time tables (256 shells == blockDim)
    s_zl[t] = __builtin_fabsf(zetas[t]) * LOG2E;
    s_ci[t] = cidx[t];
    __syncthreads();

    const int rowstep = (int)gridDim.x * ROWS_PER_TILE;

    for (int n0 = (int)blockIdx.x * ROWS_PER_TILE; n0 < nrows; n0 += rowstep) {
        const int base = n0 * 64;      // flat (row, center) index of tile start

        // ---- stage: .w components -> LDS (coalesced, prefetch next tile)
        #pragma unroll
        for (int k = 0; k < 4; ++k) {
            const int li = t + k * TPB;            // 0..1023
            const int gi = base + li;              // (n0 + li/64)*64 + (li%64)
            // prefetch next grid-stride tile (speculative, line-granular)
            __builtin_prefetch(&diffs[(gi + rowstep * 64) * 4 + 3], 0, 0);
            const int n = n0 + (li >> 6);
            if (n < nrows) {
#if USE_ASYNC_LDS
                // memory -> LDS directly, no VGPR round trip (ASYNCcnt)
                __builtin_amdgcn_global_load_async_to_lds_b32(
                    (gptr_i32)(diffs + gi * 4 + 3),
                    (lptr_i32)(s_w + li),
                    0, 0);
#else
                s_w[li] = __builtin_nontemporal_load(&diffs[gi * 4 + 3]);
#endif
            } else {
                s_w[li] = 0.0f;
            }
        }
#if USE_ASYNC_LDS
#if __has_builtin(__builtin_amdgcn_s_wait_asynccnt)
        __builtin_amdgcn_s_wait_asynccnt(0);
#else
        asm volatile("s_wait_asynccnt 0x0" ::: "memory");
#endif
#endif
        __syncthreads();

        // ---- compute: 4 shells (one center-group) per thread per sub-iter
        #pragma unroll
        for (int k = 0; k < 4; ++k) {
            const int li  = t + k * TPB;           // 0..1023
            const int n   = n0 + (li >> 6);
            const int grp = li & 63;               // shell group = 4 shells
            if (n < nrows) {
                const v4f zl4 = ((const v4f*)s_zl)[grp];
                const int s   = grp * 4;
                const int rb  = li & ~63;          // this row's w-table base
                const float r0 = __builtin_amdgcn_sqrtf(s_w[rb + s_ci[s + 0]]);
                const float r1 = __builtin_amdgcn_sqrtf(s_w[rb + s_ci[s + 1]]);
                const float r2 = __builtin_amdgcn_sqrtf(s_w[rb + s_ci[s + 2]]);
                const float r3 = __builtin_amdgcn_sqrtf(s_w[rb + s_ci[s + 3]]);
                v4f o;
                o.x = __builtin_amdgcn_exp2f(-zl4.x * r0);   // v_exp_f32
                o.y = __builtin_amdgcn_exp2f(-zl4.y * r1);
                o.z = __builtin_amdgcn_exp2f(-zl4.z * r2);
                o.w = __builtin_amdgcn_exp2f(-zl4.w * r3);
                // [N,256] fp32 -> float4 slot (n*64 + grp), coalesced b128 NT
                __builtin_nontemporal_store(o, (v4f*)out + (n * 64 + grp));
            }
        }
        __syncthreads();
    }
}

extern "C" void kernel_launch(void* const* d_in, const int* in_sizes, int n_in,
                              void* d_out, int out_size, void* d_ws, size_t ws_size,
                              hipStream_t stream) {
    (void)n_in; (void)out_size; (void)d_ws; (void)ws_size;
    const float* diffs = (const float*)d_in[0];   // [N, 64, 4]
    const float* zetas = (const float*)d_in[1];   // [256]
    const int*   cidx  = (const int*)d_in[2];     // [256]
    float*       out   = (float*)d_out;           // [N, 256]

    const int nrows = in_sizes[0] / 256;          // N = elems / (64*4)

    int blocks = 2048;                            // persistent-style grid
    const int maxb = (nrows + ROWS_PER_TILE - 1) / ROWS_PER_TILE;
    if (blocks > maxb) blocks = maxb;
    if (blocks < 1) blocks = 1;

    eebasis_kernel<<<blocks, TPB, 0, stream>>>(diffs, zetas, cidx, out, nrows);
}